// SplitPooling_75136157876558
// MI455X (gfx1250) — compile-verified
//
#include <hip/hip_runtime.h>
#include <stdint.h>

#define OUTB   7
#define CPB    16      // channels staged per TDM tile
#define NCHUNK 8       // chunks per block (pipelined, double-buffered)
#define CTOT   256
#define FH     64
#define FW     64

typedef __attribute__((ext_vector_type(4))) unsigned int su4;
typedef __attribute__((ext_vector_type(8))) unsigned int su8;

// One block per (batch, roi, 128-channel slab). The block walks NCHUNK=8
// chunks of CPB=16 channels, double-buffering the ROI's bounding tile
// [CPB x RH x RW] through LDS with the CDNA5 Tensor Data Mover:
//   issue TDM(k+1) -> s_wait_tensorcnt 1 (TDM in-order => chunk k done)
//   -> barrier -> pool chunk k from LDS.
template <int K, int TILE>
__global__ __launch_bounds__(256)
void roi_pool_tdm(const float* __restrict__ feat,
                  const float* __restrict__ rois,
                  float* __restrict__ out,
                  int nroi)
{
    __shared__ float tile[2][CPB * TILE * TILE];

    const int b     = blockIdx.z;
    const int n     = blockIdx.y;
    const int cbase = blockIdx.x * (CPB * NCHUNK);

    // ---- ROI geometry (block-uniform), exactly matching the reference ----
    const float r0 = rois[4 * n + 0];
    const float r1 = rois[4 * n + 1];
    const float r2 = rois[4 * n + 2];
    const float r3 = rois[4 * n + 3];
    const float x1f = rintf(r0 * 0.0625f);   // jnp.round = round-half-even
    const float y1f = rintf(r1 * 0.0625f);
    const float x2f = rintf(r2 * 0.0625f);
    const float y2f = rintf(r3 * 0.0625f);
    const float roiw = fmaxf(x2f - x1f + 1.0f, 1.0f);
    const float roih = fmaxf(y2f - y1f + 1.0f, 1.0f);
    const float bw = roiw * (1.0f / 7.0f);
    const float bh = roih * (1.0f / 7.0f);

    // Bounding region of all bins: rows [hs(0), he(6)), cols [ws(0), we(6))
    int h0 = (int)fminf(fmaxf(y1f, 0.0f), (float)FH);
    int hN = (int)fminf(fmaxf(ceilf(7.0f * bh) + y1f, 0.0f), (float)FH);
    int w0 = (int)fminf(fmaxf(x1f, 0.0f), (float)FW);
    int wN = (int)fminf(fmaxf(ceilf(7.0f * bw) + x1f, 0.0f), (float)FW);

    int RH = hN - h0; if (RH < 0) RH = 0; if (RH > TILE) RH = TILE;
    int RW = wN - w0; if (RW < 0) RW = 0; if (RW > TILE) RW = TILE;

    // Force descriptor ingredients into SGPRs (values are uniform).
    h0 = __builtin_amdgcn_readfirstlane(h0);
    w0 = __builtin_amdgcn_readfirstlane(w0);
    RH = __builtin_amdgcn_readfirstlane(RH);
    RW = __builtin_amdgcn_readfirstlane(RW);

    // ---- Loop-invariant D# groups 1..3 ----
    const unsigned uRW = (unsigned)RW, uRH = (unsigned)RH;
    su8 g1; su4 g2; su4 g3;
    g1[0] = (2u << 16);                                   // data_size = 2 (4 bytes)
    g1[1] = (uRW & 0xFFFFu) << 16;                        // tensor_dim0 [15:0]
    g1[2] = (uRW >> 16) | ((uRH & 0xFFFFu) << 16);        // tdim0 hi | tdim1 lo
    g1[3] = (uRH >> 16) | (uRW << 16);                    // tdim1 hi | tile_dim0
    g1[4] = uRH | ((unsigned)CPB << 16);                  // tile_dim1 | tile_dim2
    g1[5] = (unsigned)FW;                                 // tensor_dim0_stride lo32
    g1[6] = ((unsigned)(FH * FW) & 0xFFFFu) << 16;        // stride0 hi16(=0) | stride1 lo16
    g1[7] = (unsigned)(FH * FW) >> 16;                    // stride1 hi32 (=0)
    g2[0] = (unsigned)CPB;                                // tensor_dim2
    g2[1] = 1u;                                           // tensor_dim3 (tile_dim3=0)
    g2[2] = (unsigned)(FH * FW);                          // tensor_dim2_stride lo32
    g2[3] = 0u;
    g3[0] = 0u; g3[1] = 0u; g3[2] = 0u; g3[3] = 0u;

    const unsigned ldsbase = (unsigned)(unsigned long long)(uintptr_t)&tile[0][0];
    const unsigned long long gabase =
        (unsigned long long)(uintptr_t)feat +
        4ull * ((unsigned long long)(b * CTOT + cbase) * (unsigned long long)(FH * FW) +
                (unsigned long long)(h0 * FW + w0));

    const bool issuer = (threadIdx.x < 32) && (RH > 0) && (RW > 0);

    auto issue = [&](int chunk, int bufsel) {
        if (issuer) {
            const unsigned long long ga =
                gabase + (unsigned long long)chunk * (4ull * CPB * FH * FW);
            su4 g0;
            g0[0] = 1u;                                            // count=1
            g0[1] = ldsbase + (unsigned)bufsel * (CPB * TILE * TILE * 4u);
            g0[2] = (unsigned)ga;
            g0[3] = ((unsigned)(ga >> 32) & 0x01FFFFFFu) | (2u << 30);  // type=2
            asm volatile("tensor_load_to_lds %0, %1, %2, %3"
                         :
                         : "s"(g0), "s"(g1), "s"(g2), "s"(g3)
                         : "memory");
        }
    };

    // ---- Per-thread bin windows (chunk-invariant): <=4 slots/thread ----
    constexpr int NSLOT = (CPB * OUTB * OUTB + 255) / 256;   // 4 (784 outputs/chunk)
    int  srs[NSLOT], sre[NSLOT], scs[NSLOT], sce[NSLOT], sldb[NSLOT], sout[NSLOT];
    bool sok[NSLOT];
#pragma unroll
    for (int s = 0; s < NSLOT; ++s) {
        const int idx = (int)threadIdx.x + s * 256;
        sok[s] = idx < CPB * OUTB * OUTB;
        const int c   = idx / (OUTB * OUTB);
        const int bin = idx - c * (OUTB * OUTB);
        const int ph  = bin / OUTB;
        const int pw  = bin - ph * OUTB;

        int rs = (int)fminf(fmaxf(floorf((float)ph * bh) + y1f, 0.0f), (float)FH);
        int re = (int)fminf(fmaxf(ceilf((float)(ph + 1) * bh) + y1f, 0.0f), (float)FH);
        int cs = (int)fminf(fmaxf(floorf((float)pw * bw) + x1f, 0.0f), (float)FW);
        int ce = (int)fminf(fmaxf(ceilf((float)(pw + 1) * bw) + x1f, 0.0f), (float)FW);
        if (re > rs + K) re = rs + K;          // reference's K-bounded window
        if (ce > cs + K) ce = cs + K;
        if (re > h0 + RH) re = h0 + RH;        // defensive clamp into staged tile
        if (ce > w0 + RW) ce = w0 + RW;
        srs[s] = rs; sre[s] = re; scs[s] = cs; sce[s] = ce;
        sldb[s] = c * RH * RW;                 // LDS channel base (elements)
        sout[s] = c * (OUTB * OUTB) + bin;     // output offset within chunk
    }

    const size_t obase =
        ((size_t)b * (size_t)nroi + (size_t)n) * (size_t)(CTOT * OUTB * OUTB) +
        (size_t)cbase * (OUTB * OUTB);

    // ---- Pipelined chunk loop: TDM(k+1) overlaps pooling of chunk k ----
    issue(0, 0);
    for (int k = 0; k < NCHUNK; ++k) {
        if (k + 1 < NCHUNK) {
            issue(k + 1, (k + 1) & 1);
            if (threadIdx.x < 32) __builtin_amdgcn_s_wait_tensorcnt(1);
        } else {
            if (threadIdx.x < 32) __builtin_amdgcn_s_wait_tensorcnt(0);
        }
        __syncthreads();   // chunk k resident in tile[k&1] for all waves

        const float* __restrict__ buf = &tile[k & 1][0];
        const size_t ochunk = obase + (size_t)(k * CPB) * (OUTB * OUTB);
#pragma unroll
        for (int s = 0; s < NSLOT; ++s) {
            if (!sok[s]) continue;
            float v = 0.0f;                    // empty bin -> 0 (matches reference)
            if (sre[s] > srs[s] && sce[s] > scs[s]) {
                float m = -__builtin_huge_valf();
                const float* cb = buf + sldb[s];
                for (int y = srs[s]; y < sre[s]; ++y) {
                    const float* row = cb + (y - h0) * RW;
                    for (int x = scs[s]; x < sce[s]; ++x)
                        m = fmaxf(m, row[x - w0]);
                }
                v = m;
            }
            out[ochunk + (size_t)sout[s]] = v;
        }
        __syncthreads();   // all reads of tile[k&1] done before TDM reuses it
    }
}

extern "C" void kernel_launch(void* const* d_in, const int* in_sizes, int n_in,
                              void* d_out, int out_size, void* d_ws, size_t ws_size,
                              hipStream_t stream)
{
    (void)d_ws; (void)ws_size; (void)out_size;
    if (n_in < 4) return;

    const float* feat  = (const float*)d_in[0];
    const float* roisl = (const float*)d_in[1];
    const float* roism = (const float*)d_in[2];
    const float* roiss = (const float*)d_in[3];
    float* out = (float*)d_out;

    const int B  = in_sizes[0] / (CTOT * FH * FW);
    const int nl = in_sizes[1] / 4;
    const int nm = in_sizes[2] / 4;
    const int ns = in_sizes[3] / 4;

    const size_t offm = (size_t)B * (size_t)nl * (size_t)(CTOT * OUTB * OUTB);
    const size_t offs = offm + (size_t)B * (size_t)nm * (size_t)(CTOT * OUTB * OUTB);

    const int gx = CTOT / (CPB * NCHUNK);   // 2 channel slabs per (b, roi)
    dim3 blk(256, 1, 1);
    if (nl > 0)
        roi_pool_tdm<6, 26><<<dim3(gx, nl, B), blk, 0, stream>>>(feat, roisl, out, nl);
    if (nm > 0)
        roi_pool_tdm<5, 18><<<dim3(gx, nm, B), blk, 0, stream>>>(feat, roism, out + offm, nm);
    if (ns > 0)
        roi_pool_tdm<4, 12><<<dim3(gx, ns, B), blk, 0, stream>>>(feat, roiss, out + offs, ns);
}